// CRF_26955214750087
// MI455X (gfx1250) — compile-verified
//
#include <hip/hip_runtime.h>
#include <hip/hip_bf16.h>
#include <stdint.h>

// Problem sizes (fixed by the reference)
#define BB 128
#define TT 1024
#define DD 256
#define UU 48

#define KT       32              // K-tile staged per double-buffer step
#define XS       36              // padded LDS row stride (floats): conflict-free A reads
#define NKT      (DD / KT)       // 8 K-tiles

typedef __attribute__((ext_vector_type(2))) float v2f;
typedef __attribute__((ext_vector_type(8))) float v8f;

// GVS-mode async DMA: copies 16 bytes from the global address sbase + voff to the LDS address in lds_off, ASYNCcnt-tracked.
__device__ __forceinline__ void async_ld_b128(uint32_t lds_off, const float* sbase,
                                              uint32_t voff) {
    asm volatile("global_load_async_to_lds_b128 %0, %1, %2"
                 :
                 : "v"(lds_off), "v"(voff), "s"(sbase)
                 : "memory");
}
__device__ __forceinline__ void wait_async0() {
    asm volatile("s_wait_asynccnt 0x0" ::: "memory");
}

// ---------------------------------------------------------------------------
// Kernel 1: logits[M, 48] = X[M, 256] * W[256, 48] + bias   (M = B*T = 131072)
//
// Block = 256 threads = 8 waves; each wave owns a 16-row M-tile, all 3 N-tiles.
// W is staged once, pre-swizzled into the B-fragment layout so every B operand
// is one aligned ds_load_b64 (no register shuffles, banks conflict-free).
// X streams through a double-buffered LDS tile filled by
// GLOBAL_LOAD_ASYNC_TO_LDS_B128 (coalesced: wave request = 4 full cache lines),
// overlapped with the WMMA compute on the other buffer.
// ---------------------------------------------------------------------------
__global__ __launch_bounds__(256) void crf_gemm_logits(
    const float* __restrict__ X, const float* __restrict__ W,
    const float* __restrict__ bias, float* __restrict__ L)
{
    // B-fragment swizzle: w_sh[p*96 + c*2 + r] = W[2p + r, c]  (p = K-pair)
    __shared__ __align__(16) float w_sh[DD / 2 * UU * 2];        // 48 KB
    __shared__ __align__(16) float x_sh[2][128 * XS];            // 36 KB

    const int tid = threadIdx.x;
    const long mBlock = (long)blockIdx.x * 128;

    // ---- stage W, swizzled (one time, coalesced global reads) ----
    for (int i = tid; i < DD * UU; i += 256) {
        const int d = i / UU;                 // K index 0..255
        const int c = i - d * UU;             // col 0..47
        w_sh[(d >> 1) * (UU * 2) + c * 2 + (d & 1)] = W[i];
    }

    // ---- async-stage first X K-tile ----
    // tile chunk ci (0..1023): row = ci>>3, kk = (ci&7)*4  -> b128 each
    const uint32_t xbase0 = (uint32_t)(uintptr_t)&x_sh[0][0];
    const uint32_t xbase1 = (uint32_t)(uintptr_t)&x_sh[1][0];
    #pragma unroll
    for (int j = 0; j < 4; ++j) {
        const int ci  = tid + j * 256;
        const int row = ci >> 3;
        const int kk  = (ci & 7) * 4;
        async_ld_b128(xbase0 + (uint32_t)(row * XS + kk) * 4u, X,
                      (uint32_t)((mBlock + row) * DD + kk) * 4u);
    }
    wait_async0();
    __syncthreads();

    const int wave = tid >> 5;
    const int lane = tid & 31;
    const int half = lane >> 4;               // lane half -> K pair {2h, 2h+1}
    const int l16  = lane & 15;
    const int rowL = wave * 16 + l16;         // local A row 0..127

    v8f acc[3];
    #pragma unroll
    for (int n = 0; n < 3; ++n) {
        const float bv = bias[n * 16 + l16];
        acc[n] = (v8f){bv, bv, bv, bv, bv, bv, bv, bv};
    }

    for (int kt = 0; kt < NKT; ++kt) {
        const int buf = kt & 1;

        // prefetch next K-tile into the other buffer (async LDS DMA)
        if (kt + 1 < NKT) {
            const uint32_t xb = (buf ? xbase0 : xbase1);
            #pragma unroll
            for (int j = 0; j < 4; ++j) {
                const int ci  = tid + j * 256;
                const int row = ci >> 3;
                const int kk  = (ci & 7) * 4;
                async_ld_b128(xb + (uint32_t)(row * XS + kk) * 4u, X,
                              (uint32_t)((mBlock + row) * DD + (kt + 1) * KT + kk) * 4u);
            }
        }

        const float* __restrict__ xs = &x_sh[buf][rowL * XS + 2 * half];
        const float* __restrict__ ws = &w_sh[(kt * (KT / 2) + half) * (UU * 2) + l16 * 2];

        #pragma unroll
        for (int k = 0; k < KT; k += 4) {
            // A fragment: lanes 0-15 K = k,k+1 ; lanes 16-31 K = k+2,k+3
            const v2f a = *(const v2f*)(xs + k);                  // ds_load_b64
            #pragma unroll
            for (int n = 0; n < 3; ++n) {
                const v2f b = *(const v2f*)(ws + (k / 2) * (UU * 2) + n * 32);
                acc[n] = __builtin_amdgcn_wmma_f32_16x16x4_f32(
                    false, a, false, b, (short)0, acc[n], false, false);
            }
        }

        wait_async0();        // next tile landed in LDS (this wave's DMAs)
        __syncthreads();      // ... and visible to all waves
    }

    // C/D layout: VGPR r, lanes 0-15 -> M = r, lanes 16-31 -> M = r + 8
    const long mBase = mBlock + wave * 16;
    #pragma unroll
    for (int n = 0; n < 3; ++n) {
        #pragma unroll
        for (int r = 0; r < 8; ++r) {
            const long row = mBase + r + 8 * half;
            L[row * UU + n * 16 + l16] = acc[n][r];
        }
    }
}

// ---------------------------------------------------------------------------
// Kernel 2: Viterbi forward + backtrace, one workgroup per batch element.
// trans (9 KB), double-buffered state and ALL back-pointers (~48 KB) live in
// LDS (~58 KB/block, fits the 320 KB WGP LDS); backtrace never touches HBM.
// Lane j owns state j; strict '>' keeps jnp.argmax first-max tie-breaking.
// ---------------------------------------------------------------------------
__global__ __launch_bounds__(64) void crf_viterbi(
    const float* __restrict__ L, const float* __restrict__ trans,
    float* __restrict__ out)
{
    __shared__ float          tr[UU * UU];        // 9216 B
    __shared__ float          st[2][UU];          //  384 B
    __shared__ unsigned char  bp[(TT - 1) * UU];  // 49104 B

    const int b   = blockIdx.x;
    const int tid = threadIdx.x;
    const float* __restrict__ lb = L + (long)b * TT * UU;

    for (int i = tid; i < UU * UU; i += 64) tr[i] = trans[i];
    if (tid < UU) st[0][tid] = lb[tid];           // init_state = logits[:, 0]
    __syncthreads();

    int cur = 0;
    for (int t = 1; t < TT; ++t) {
        if (tid < UU) {
            float m  = -3.402823466e38f;
            int   bi = 0;
            #pragma unroll 8
            for (int i = 0; i < UU; ++i) {
                const float s = st[cur][i] + tr[i * UU + tid];
                if (s > m) { m = s; bi = i; }
            }
            st[cur ^ 1][tid]       = lb[t * UU + tid] + m;
            bp[(t - 1) * UU + tid] = (unsigned char)bi;
        }
        __syncthreads();
        cur ^= 1;
    }

    if (tid == 0) {
        int   tag  = 0;
        float best = st[cur][0];
        for (int j = 1; j < UU; ++j) {
            const float v = st[cur][j];
            if (v > best) { best = v; tag = j; }
        }
        float* __restrict__ ob = out + (long)b * TT;
        ob[TT - 1] = (float)tag;
        for (int t = TT - 2; t >= 0; --t) {
            tag   = bp[t * UU + tag];
            ob[t] = (float)tag;
        }
    }
}

// ---------------------------------------------------------------------------
extern "C" void kernel_launch(void* const* d_in, const int* in_sizes, int n_in,
                              void* d_out, int out_size, void* d_ws, size_t ws_size,
                              hipStream_t stream)
{
    const float* x     = (const float*)d_in[0];  // (B, T, D) fp32
    const float* kern  = (const float*)d_in[1];  // (D, U)    fp32
    const float* bias  = (const float*)d_in[2];  // (U,)      fp32
    const float* chain = (const float*)d_in[3];  // (U, U)    fp32

    float* logits = (float*)d_ws;                // B*T*U fp32 = 24 MB scratch
    float* out    = (float*)d_out;               // (B, T) fp32

    crf_gemm_logits<<<1024, 256, 0, stream>>>(x, kern, bias, logits);
    crf_viterbi<<<BB, 64, 0, stream>>>(logits, chain, out);
}